// Constant_Pool_layer_60842506715664
// MI455X (gfx1250) — compile-verified
//
#include <hip/hip_runtime.h>
#include <hip/hip_bf16.h>
#include <float.h>

typedef __attribute__((ext_vector_type(2))) float    v2f;
typedef __attribute__((ext_vector_type(8))) float    v8f;
typedef __attribute__((ext_vector_type(4))) unsigned v4u;
typedef __attribute__((ext_vector_type(8))) int      v8i;
typedef __attribute__((ext_vector_type(4))) int      v4i;

#define NB    8      // batches
#define NV    4096   // vertices
#define NC    256    // channels
#define KNN   16
#define POOL  1024

// ---------------------------------------------------------------------------
// JAX threefry2x32 (exact algorithm: 20 rounds, rotation schedule, key sched)
// ---------------------------------------------------------------------------
__device__ __forceinline__ unsigned rotl32(unsigned v, int r) {
    return (v << r) | (v >> (32 - r));
}

__device__ void threefry2x32(unsigned k0, unsigned k1, unsigned c0, unsigned c1,
                             unsigned& o0, unsigned& o1) {
    const unsigned ks2 = k0 ^ k1 ^ 0x1BD11BDAu;
    unsigned x0 = c0 + k0, x1 = c1 + k1;
    const int ra[4] = {13, 15, 26, 6};
    const int rb[4] = {17, 29, 16, 24};
#pragma unroll
    for (int i = 0; i < 4; ++i) { x0 += x1; x1 = rotl32(x1, ra[i]); x1 ^= x0; }
    x0 += k1;  x1 += ks2 + 1u;
#pragma unroll
    for (int i = 0; i < 4; ++i) { x0 += x1; x1 = rotl32(x1, rb[i]); x1 ^= x0; }
    x0 += ks2; x1 += k0 + 2u;
#pragma unroll
    for (int i = 0; i < 4; ++i) { x0 += x1; x1 = rotl32(x1, ra[i]); x1 ^= x0; }
    x0 += k0;  x1 += k1 + 3u;
#pragma unroll
    for (int i = 0; i < 4; ++i) { x0 += x1; x1 = rotl32(x1, rb[i]); x1 ^= x0; }
    x0 += k1;  x1 += ks2 + 4u;
#pragma unroll
    for (int i = 0; i < 4; ++i) { x0 += x1; x1 = rotl32(x1, ra[i]); x1 ^= x0; }
    x0 += ks2; x1 += k0 + 5u;
    o0 = x0; o1 = x1;
}

__device__ __forceinline__ unsigned jax_bits4096(unsigned s0, unsigned s1, int i) {
    unsigned p = (unsigned)(i & 2047);
    unsigned o0, o1;
    threefry2x32(s0, s1, p, p + 2048u, o0, o1);
    return (i < 2048) ? o0 : o1;
}

// ---------------------------------------------------------------------------
// Kernel A: replicate jax.random.permutation(key(42), 4096)[:1024].
// 2 rounds of stable sort-by-random-key; stability via 56-bit composite
// (sortkey<<24 | pos<<12 | val) bitonic-sorted in LDS. Also emits the
// gathered vertices_pool output.
// ---------------------------------------------------------------------------
__global__ __launch_bounds__(1024) void perm_kernel(const float* __restrict__ vtx,
                                                    float* __restrict__ out_v,
                                                    int* __restrict__ sidx) {
    __shared__ unsigned long long comp[NV];
    const unsigned tid = threadIdx.x;

    unsigned k0 = 0u, k1 = 42u;  // jax.random.key(42) -> (0, 42)

    for (int round = 0; round < 2; ++round) {
        unsigned a0, a1, b0, b1;
        threefry2x32(k0, k1, 0u, 2u, a0, a1);
        threefry2x32(k0, k1, 1u, 3u, b0, b1);
        const unsigned s0 = a1, s1 = b1;  // subkey = (bits[2], bits[3])
        k0 = a0; k1 = b0;                 // key    = (bits[0], bits[1])

        for (int i = tid; i < NV; i += blockDim.x) {
            unsigned sk  = jax_bits4096(s0, s1, i);
            unsigned val = (round == 0) ? (unsigned)i
                                        : (unsigned)(comp[i] & 0xFFFu);
            comp[i] = ((unsigned long long)sk << 24) |
                      ((unsigned long long)(unsigned)i << 12) |
                      (unsigned long long)val;
        }
        __syncthreads();

        for (unsigned k = 2; k <= NV; k <<= 1) {
            for (unsigned j = k >> 1; j > 0; j >>= 1) {
                for (unsigned i = tid; i < NV; i += blockDim.x) {
                    unsigned ixj = i ^ j;
                    if (ixj > i) {
                        bool up = ((i & k) == 0);
                        unsigned long long a = comp[i], c = comp[ixj];
                        bool sw = up ? (a > c) : (a < c);
                        if (sw) { comp[i] = c; comp[ixj] = a; }
                    }
                }
                __syncthreads();
            }
        }
    }

    for (int i = tid; i < POOL; i += blockDim.x)
        sidx[i] = (int)(comp[i] & 0xFFFu);

    for (int i = tid; i < NB * POOL; i += blockDim.x) {
        int b = i / POOL, q = i - b * POOL;
        int s = (int)(comp[q] & 0xFFFu);
        const float* src = vtx + ((size_t)b * NV + s) * 3;
        float* dst = out_v + (size_t)i * 3;
        dst[0] = src[0]; dst[1] = src[1]; dst[2] = src[2];
    }
}

// ---------------------------------------------------------------------------
// Kernel B: TDM-staged WMMA kNN + neighbor-feature max-pool.
//
// All 8 waves of a block share one batch. Wave 0 DMAs the batch's raw
// vertex array (4096*3 f32 = 48 KB, contiguous) into LDS with one
// TENSOR_LOAD_TO_LDS (1-D descriptor), s_wait_tensorcnt 0, barrier; the
// block then reformats it into the WMMA A-operand table (x,y,z,|c|^2).
// Inner loop (software-pipelined): one ds_load_b64 for the NEXT tile's A
// fragment overlaps the current tile's V_WMMA_F32_16X16X4_F32 + top-17
// insertion.  A = candidates (cx,cy,cz,|c|^2), B = queries
// (-2qx,-2qy,-2qz,1) -> D[M=cand][N=query] = |c|^2 - 2 q.c (row-constant
// |q|^2 dropped: ordering/tie-breaks identical to reference).  D layout
// puts query n on lanes n/n+16 -> per-lane register top-17, LDS merge of
// the two half-lists; merge scratch aliases the dead candidate table.
// ---------------------------------------------------------------------------
__global__ __launch_bounds__(256) void knn_pool_kernel(const float* __restrict__ vtx,
                                                       const float* __restrict__ feat,
                                                       const int* __restrict__ sidx,
                                                       float* __restrict__ out_f) {
    __shared__ __align__(16) unsigned char smem[NV * 16 + NV * 12];  // 64K + 48K

    const int lane = threadIdx.x & 31;
    const int wave = threadIdx.x >> 5;
    const int tile = blockIdx.x * 8 + wave;   // 512 tiles = 8 batches * 64
    const int b    = tile >> 6;               // same for all waves in block
    const int tq0  = (tile & 63) * 16;        // first sampled query in batch

    const int col = lane & 15;
    const int hi  = lane >> 4;

    // ---- phase 0: TDM the raw vertex block for this batch into LDS ----
    if (wave == 0) {
        unsigned long long ga =
            (unsigned long long)(uintptr_t)(vtx + (size_t)b * NV * 3);
        unsigned lds_raw = (unsigned)(uintptr_t)(&smem[NV * 16]);

        v4u g0;
        g0[0] = 1u;                                   // count=1, user mode
        g0[1] = lds_raw;                              // lds_addr (bytes)
        g0[2] = (unsigned)(ga & 0xFFFFFFFFu);         // global_addr[31:0]
        g0[3] = (unsigned)((ga >> 32) & 0x01FFFFFFu)  // global_addr[56:32]
                | (2u << 30);                         // type = 2 ("image")

        const unsigned n0 = NV * 3;                   // 12288 f32 elements
        v8i g1;
        g1[0] = (int)(2u << 16);                      // data_size = 4 bytes
        g1[1] = (int)((n0 & 0xFFFFu) << 16);          // tensor_dim0[15:0]
        g1[2] = (int)((n0 >> 16) | (1u << 16));       // dim0 hi | tensor_dim1=1
        g1[3] = (int)((n0 & 0xFFFFu) << 16);          // tile_dim0 = 12288
        g1[4] = 1;                                    // tile_dim1=1, tile_dim2=0
        g1[5] = (int)n0;                              // tensor_dim0_stride lo
        g1[6] = 0;                                    // stride hi | dim1_stride lo
        g1[7] = 0;                                    // tensor_dim1_stride hi

        v4i g2 = {0, 0, 0, 0};
        v4i g3 = {0, 0, 0, 0};
        v8i g4 = {0, 0, 0, 0, 0, 0, 0, 0};
        __builtin_amdgcn_tensor_load_to_lds(g0, g1, g2, g3, g4, 0);
        __builtin_amdgcn_s_wait_tensorcnt(0);
    }
    __syncthreads();

    // ---- phase 1: reformat raw xyz -> A-operand table (x,y,z,|c|^2) ----
    {
        const float* raw = (const float*)(smem + NV * 16);
        float4* cand = (float4*)smem;
        for (int v = threadIdx.x; v < NV; v += 256) {
            float x = raw[v * 3 + 0], y = raw[v * 3 + 1], z = raw[v * 3 + 2];
            float4 e; e.x = x; e.y = y; e.z = z; e.w = x * x + y * y + z * z;
            cand[v] = e;
        }
    }

    // B operand (loop-invariant): query of column `col`
    const int qv = sidx[tq0 + col];
    const float* qp = vtx + ((size_t)b * NV + qv) * 3;
    const float qx = qp[0], qy = qp[1], qz = qp[2];
    v2f bmat;
    bmat.x = (hi == 0) ? -2.0f * qx : -2.0f * qz;
    bmat.y = (hi == 0) ? -2.0f * qy : 1.0f;

    // per-lane sorted top-17 (ascending distance, stable lower-index ties)
    float bd[17]; int bi[17];
#pragma unroll
    for (int k = 0; k < 17; ++k) { bd[k] = FLT_MAX; bi[k] = 0x7FFFFFFF; }

    __syncthreads();

    const float2* candh = (const float2*)smem;  // cand[v].xy at 2v, .zw at 2v+1
    const int aoff = 2 * col + hi;              // lane's fragment within a tile

    // software pipeline: fetch next tile's fragment before consuming current
    float2 af = candh[aoff];
#pragma unroll 2
    for (int j = 0; j < NV / 16; ++j) {
        float2 nx = candh[(((j + 1) & 255) * 32) + aoff];  // ds_load_b64 (next)

        v2f amat; amat.x = af.x; amat.y = af.y;
        v8f acc = {};
        acc = __builtin_amdgcn_wmma_f32_16x16x4_f32(
            false, amat, false, bmat, (short)0, acc, false, false);

#pragma unroll
        for (int r = 0; r < 8; ++r) {
            float dist = acc[r];
            int   idx  = j * 16 + r + hi * 8;
            if (dist < bd[16]) {  // strict: equal dist keeps lower index
                bd[16] = dist; bi[16] = idx;
#pragma unroll
                for (int k = 16; k > 0; --k) {
                    if (bd[k] < bd[k - 1]) {
                        float td = bd[k]; bd[k] = bd[k - 1]; bd[k - 1] = td;
                        int   ti = bi[k]; bi[k] = bi[k - 1]; bi[k - 1] = ti;
                    }
                }
            }
        }
        af = nx;
    }

    __syncthreads();  // candidate table dead; alias merge scratch over it

    float* md  = (float*)smem;                  // [8][16][34] = 17408 B
    int*   mi  = (int*)(smem + 17408);          // [8][16][34] = 17408 B
    int*   nbr = (int*)(smem + 34816);          // [8][16][16] =  8192 B

#pragma unroll
    for (int k = 0; k < 17; ++k) {
        md[(wave * 16 + col) * 34 + hi * 17 + k] = bd[k];
        mi[(wave * 16 + col) * 34 + hi * 17 + k] = bi[k];
    }
    __syncthreads();

    if (hi == 0) {
        const float* mdq = md + (wave * 16 + col) * 34;
        const int*   miq = mi + (wave * 16 + col) * 34;
        int p0 = 0, p1 = 0;
        for (int t = 0; t < 17; ++t) {
            float da = mdq[p0];      int ia = miq[p0];
            float db = mdq[17 + p1]; int ib = miq[17 + p1];
            bool takeA = (da < db) || (da == db && ia < ib);
            int sel = takeA ? ia : ib;
            if (takeA) ++p0; else ++p1;
            if (t > 0) nbr[(wave * 16 + col) * 16 + (t - 1)] = sel;
        }
    }
    __syncthreads();

    // max-pool 16 neighbor feature rows; lane covers 8 channels (2x float4);
    // prefetch next query's neighbor rows (global_prefetch_b8).
    const float* fb = feat + (size_t)b * NV * NC;
    for (int q = 0; q < 16; ++q) {
        if (q < 15) {
            int pn = nbr[(wave * 16 + q + 1) * 16 + col];
            __builtin_prefetch(fb + (size_t)pn * NC + hi * 32, 0, 1);
        }
        const int ch = lane * 8;
        int nb0 = nbr[(wave * 16 + q) * 16 + 0];
        const float* p0 = fb + (size_t)nb0 * NC + ch;
        float4 m0 = *(const float4*)p0;
        float4 m1 = *(const float4*)(p0 + 4);
#pragma unroll 5
        for (int n = 1; n < KNN; ++n) {
            int nb = nbr[(wave * 16 + q) * 16 + n];
            const float* pp = fb + (size_t)nb * NC + ch;
            float4 a0 = *(const float4*)pp;
            float4 a1 = *(const float4*)(pp + 4);
            m0.x = fmaxf(m0.x, a0.x); m0.y = fmaxf(m0.y, a0.y);
            m0.z = fmaxf(m0.z, a0.z); m0.w = fmaxf(m0.w, a0.w);
            m1.x = fmaxf(m1.x, a1.x); m1.y = fmaxf(m1.y, a1.y);
            m1.z = fmaxf(m1.z, a1.z); m1.w = fmaxf(m1.w, a1.w);
        }
        float* op = out_f + ((size_t)b * POOL + tq0 + q) * NC + ch;
        *(float4*)op = m0;
        *(float4*)(op + 4) = m1;
    }
}

// ---------------------------------------------------------------------------
extern "C" void kernel_launch(void* const* d_in, const int* in_sizes, int n_in,
                              void* d_out, int out_size, void* d_ws, size_t ws_size,
                              hipStream_t stream) {
    const float* vtx  = (const float*)d_in[0];  // (8,4096,3)
    const float* feat = (const float*)d_in[1];  // (8,4096,256)
    float* out   = (float*)d_out;
    float* out_v = out;                               // (8,1024,3)
    float* out_f = out + (size_t)NB * POOL * 3;       // (8,1024,256)
    int*   sidx  = (int*)d_ws;                        // 1024 ints

    perm_kernel<<<1, 1024, 0, stream>>>(vtx, out_v, sidx);
    knn_pool_kernel<<<64, 256, 0, stream>>>(vtx, feat, sidx, out_f);
}